// OnlyMeanPooledSpanEmbeddingLayer_87316685128605
// MI455X (gfx1250) — compile-verified
//
#include <hip/hip_runtime.h>
#include <stdint.h>

// Problem constants (match reference)
#define VOCAB 151936
#define HID   1536
#define BATCH 8
#define SEQ   2048
#define NSPAN 256
#define NSEG  (BATCH * NSPAN + 1)   // 2049 (last = dump bucket, never used)
#define NTOK  (BATCH * SEQ)         // 16384
#define LN_EPS 1e-5f

// ---------------------------------------------------------------------------
// Kernel 1: zero the workspace (sums[NSEG*HID] ++ counts[NSEG]) every launch.
// ---------------------------------------------------------------------------
__global__ __launch_bounds__(256) void zero_ws_kernel(float* ws, int n) {
    int i = blockIdx.x * blockDim.x + threadIdx.x;
    if (i < n) ws[i] = 0.0f;
}

// ---------------------------------------------------------------------------
// Kernel 2: segmented sum. One wave (32 lanes) per token; skip out-of-span
// tokens (their bucket is never read). Row = 12 x float4 per lane pattern:
// lane l, iter j covers float4 index (j*32 + l)  -> fully coalesced 512B/instr.
// ---------------------------------------------------------------------------
__global__ __launch_bounds__(256) void accumulate_kernel(
        const int* __restrict__ ids, const int* __restrict__ segs,
        const float* __restrict__ emb, float* __restrict__ sums,
        float* __restrict__ counts) {
    int wave = (blockIdx.x * blockDim.x + threadIdx.x) >> 5;   // token id
    int lane = threadIdx.x & 31;
    if (wave >= NTOK) return;
    int t   = wave;
    int sid = segs[t];
    if (sid >= NSPAN) return;                 // wave-uniform: token not in a span
    int seg = (t >> 11) * NSPAN + sid;        // t/SEQ * NSPAN + sid
    if (lane == 0) atomicAdd(counts + seg, 1.0f);
    const float4* src = (const float4*)(emb + (size_t)ids[t] * HID);
    float*        dst = sums + (size_t)seg * HID;
#pragma unroll
    for (int j = 0; j < 12; ++j) {
        float4 v = src[j * 32 + lane];
        int e = (j * 32 + lane) * 4;
        atomicAdd(dst + e + 0, v.x);
        atomicAdd(dst + e + 1, v.y);
        atomicAdd(dst + e + 2, v.z);
        atomicAdd(dst + e + 3, v.w);
    }
}

// ---------------------------------------------------------------------------
// Async global->LDS copy of one 6KB row (12 x B128 per lane, ASYNCcnt += 12).
// IOFFSET is added to BOTH the LDS and global address per the CDNA5 ISA, so a
// single (lds,global) base pair covers the row. EXEC is all-ones here
// (uniform control flow, full waves).
// ---------------------------------------------------------------------------
__device__ __forceinline__ void async_copy_row(const float* src, float* ldsbuf, int lane) {
    uint32_t la = (uint32_t)(uintptr_t)ldsbuf + (uint32_t)(lane * 16); // LDS offset = low 32 bits of the pointer
    uint64_t ga = (uint64_t)(uintptr_t)src + (uint64_t)(lane * 16);
    asm volatile(
        "global_load_async_to_lds_b128 %0, %1, off\n\t"
        "global_load_async_to_lds_b128 %0, %1, off offset:512\n\t"
        "global_load_async_to_lds_b128 %0, %1, off offset:1024\n\t"
        "global_load_async_to_lds_b128 %0, %1, off offset:1536\n\t"
        "global_load_async_to_lds_b128 %0, %1, off offset:2048\n\t"
        "global_load_async_to_lds_b128 %0, %1, off offset:2560\n\t"
        "global_load_async_to_lds_b128 %0, %1, off offset:3072\n\t"
        "global_load_async_to_lds_b128 %0, %1, off offset:3584\n\t"
        "global_load_async_to_lds_b128 %0, %1, off offset:4096\n\t"
        "global_load_async_to_lds_b128 %0, %1, off offset:4608\n\t"
        "global_load_async_to_lds_b128 %0, %1, off offset:5120\n\t"
        "global_load_async_to_lds_b128 %0, %1, off offset:5632\n\t"
        :: "v"(la), "v"(ga) : "memory");
}

// Wave-uniform source selection: pooled mean row (sums[seg], scaled by 1/cnt)
// when in-span with count>1.5, else the raw embedding row.
__device__ __forceinline__ const float* pick_src(
        int t, const int* __restrict__ ids, const int* __restrict__ segs,
        const float* __restrict__ emb, const float* __restrict__ sums,
        const float* __restrict__ counts, float* scale_out) {
    float scale = 1.0f;
    const float* src = nullptr;
    int sid = segs[t];
    if (sid < NSPAN) {
        int seg = (t >> 11) * NSPAN + sid;
        float c = counts[seg];
        if (c > 1.5f) {
            scale = 1.0f / c;
            src = sums + (size_t)seg * HID;
        }
    }
    if (src == nullptr) src = emb + (size_t)ids[t] * HID;
    *scale_out = scale;
    return src;
}

// LayerNorm of one LDS-staged row; wave32 shuffle reduction; float4 output.
__device__ __forceinline__ void process_row(
        const float* __restrict__ ldsbuf, float scale, int lane,
        const float* __restrict__ gamma, const float* __restrict__ beta,
        float* __restrict__ outrow) {
    const float4* lb = (const float4*)ldsbuf;
    float4 v[12];
    float s = 0.0f, sq = 0.0f;
#pragma unroll
    for (int j = 0; j < 12; ++j) {
        float4 x = lb[j * 32 + lane];
        x.x *= scale; x.y *= scale; x.z *= scale; x.w *= scale;
        v[j] = x;
        s  += x.x + x.y + x.z + x.w;
        sq += x.x * x.x + x.y * x.y + x.z * x.z + x.w * x.w;
    }
#pragma unroll
    for (int off = 16; off > 0; off >>= 1) {
        s  += __shfl_xor(s,  off, 32);
        sq += __shfl_xor(sq, off, 32);
    }
    const float inv = 1.0f / (float)HID;
    float mu   = s * inv;
    float var  = sq * inv - mu * mu;
    float rstd = rsqrtf(var + LN_EPS);
    const float4* g4 = (const float4*)gamma;
    const float4* b4 = (const float4*)beta;
    float4*       o4 = (float4*)outrow;
#pragma unroll
    for (int j = 0; j < 12; ++j) {
        int idx = j * 32 + lane;
        float4 g = g4[idx], b = b4[idx], x = v[j], o;
        o.x = (x.x - mu) * rstd * g.x + b.x;
        o.y = (x.y - mu) * rstd * g.y + b.y;
        o.z = (x.z - mu) * rstd * g.z + b.z;
        o.w = (x.w - mu) * rstd * g.w + b.w;
        o4[idx] = o;
    }
}

// ---------------------------------------------------------------------------
// Kernel 3: finalize. 128 threads = 4 waves; each wave owns 4 consecutive
// tokens and a 2 x 6KB double buffer in LDS (48KB/block). Prefetch token i+1
// asynchronously while LayerNorming token i; s_wait_asynccnt 12 releases the
// oldest 12 async ops (in-order completion) = the current buffer.
// ---------------------------------------------------------------------------
__global__ __launch_bounds__(128) void finalize_kernel(
        const int* __restrict__ ids, const int* __restrict__ segs,
        const float* __restrict__ emb, const float* __restrict__ gamma,
        const float* __restrict__ beta, const float* __restrict__ sums,
        const float* __restrict__ counts, float* __restrict__ out) {
    __shared__ float lds[4 * 2 * HID];         // 49152 bytes
    int wv   = threadIdx.x >> 5;
    int lane = threadIdx.x & 31;
    int w    = blockIdx.x * 4 + wv;            // global wave id
    int t0   = w * 4;                          // first of 4 tokens for this wave

    float* buf0 = &lds[wv * 2 * HID];
    float* buf1 = buf0 + HID;
    float sc[2];

    const float* src = pick_src(t0, ids, segs, emb, sums, counts, &sc[0]);
    async_copy_row(src, buf0, lane);

#pragma unroll
    for (int i = 0; i < 4; ++i) {
        int pb = i & 1;
        float* cur = pb ? buf1 : buf0;
        if (i + 1 < 4) {
            float* nxt = pb ? buf0 : buf1;
            const float* nsrc = pick_src(t0 + i + 1, ids, segs, emb, sums, counts, &sc[pb ^ 1]);
            async_copy_row(nsrc, nxt, lane);
            asm volatile("s_wait_asynccnt 0xc" ::: "memory"); // oldest 12 (current buf) done
        } else {
            asm volatile("s_wait_asynccnt 0x0" ::: "memory");
        }
        process_row(cur, sc[pb], lane, gamma, beta, out + (size_t)(t0 + i) * HID);
    }
}

// ---------------------------------------------------------------------------
extern "C" void kernel_launch(void* const* d_in, const int* in_sizes, int n_in,
                              void* d_out, int out_size, void* d_ws, size_t ws_size,
                              hipStream_t stream) {
    (void)in_sizes; (void)n_in; (void)out_size; (void)ws_size;
    const int*   input_ids   = (const int*)d_in[0];
    const int*   segment_ids = (const int*)d_in[1];
    const float* emb_table   = (const float*)d_in[2];
    const float* gamma       = (const float*)d_in[3];
    const float* beta        = (const float*)d_in[4];
    float*       out         = (float*)d_out;

    float* sums   = (float*)d_ws;                       // NSEG*HID floats
    float* counts = sums + (size_t)NSEG * HID;          // NSEG floats
    const int nzero = NSEG * HID + NSEG;

    zero_ws_kernel<<<(nzero + 255) / 256, 256, 0, stream>>>(sums, nzero);

    // one wave per token: 16384 waves = 2048 blocks of 256 threads
    accumulate_kernel<<<NTOK / 8, 256, 0, stream>>>(input_ids, segment_ids,
                                                    emb_table, sums, counts);

    // 4 waves/block * 4 tokens/wave = 16 tokens/block -> 1024 blocks
    finalize_kernel<<<NTOK / 16, 128, 0, stream>>>(input_ids, segment_ids, emb_table,
                                                   gamma, beta, sums, counts, out);
}